// AiLUT_21165598835346
// MI455X (gfx1250) — compile-verified
//
#include <hip/hip_runtime.h>
#include <hip/hip_bf16.h>

typedef float v2f __attribute__((ext_vector_type(2)));
typedef float v4f __attribute__((ext_vector_type(4)));
typedef float v8f __attribute__((ext_vector_type(8)));

#define LUT_D  33
#define LUT_D3 35937   // 33^3

// ---------------- 1. bilinear resize 1024 -> 256 ----------------
// half-pixel, scale 4: src = 4*d + 1.5 exactly => average of 2x2 at (4d+1, 4d+2)
__global__ __launch_bounds__(256) void resize4x_kernel(
    const float* __restrict__ in, float* __restrict__ out)
{
    int i = blockIdx.x * 256 + threadIdx.x;          // [0, 4*3*256*256)
    int j  = i & 255;
    int t  = i >> 8;
    int r  = t & 255;
    int bc = t >> 8;                                  // [0,12)
    const float* p = in + ((size_t)bc * 1024 + (r * 4 + 1)) * 1024 + (j * 4 + 1);
    out[i] = 0.25f * (p[0] + p[1] + p[1024] + p[1025]);
}

// ---------------- 2. stride-2 3x3 conv as implicit GEMM (WMMA f32 16x16x4) ---
// Per wave: D[16 pos x 16 oc] += A[16x4] * B[4x16]; K padded to Cin*12 so each
// input channel is exactly 3 WMMA steps with loop-invariant slot metadata.
// A layout (ISA 16x4 f32): VGPR v, lane l -> M=l%16, K=2*(l/16)+v
// B layout (mirror):       VGPR v, lane l -> N=l%16, K=2*(l/16)+v
// C/D layout:              VGPR r, lane l -> M=r+8*(l/16), N=l%16
// All layer shapes: P = Hout*Wout and Cout are multiples of 16; Wout = 2^shift.
__global__ __launch_bounds__(256) void conv_s2_wmma_kernel(
    const float* __restrict__ x, const float* __restrict__ w,
    const float* __restrict__ bias, float* __restrict__ y,
    int Cin, int Cout, int Hin, int Win, int woutShift,
    int mtiles, int ntiles)
{
    int wid = __builtin_amdgcn_readfirstlane(
        blockIdx.x * (blockDim.x >> 5) + (threadIdx.x >> 5));
    int lane = threadIdx.x & 31;

    int b  = wid / (mtiles * ntiles);
    int rr = wid % (mtiles * ntiles);
    int mt = rr / ntiles;
    int nt = rr % ntiles;

    const int Wout  = 1 << woutShift;
    const int Hout  = Hin >> 1;
    const int P     = Hout << woutShift;
    const int K9    = Cin * 9;
    const int plane = Hin * Win;

    int m  = lane & 15;
    int kp = lane >> 4;                    // 0 or 1
    int pos = mt * 16 + m;
    int oh = pos >> woutShift, ow = pos & (Wout - 1);
    int ih0 = oh * 2 - 1,      iw0 = ow * 2 - 1;
    int oc  = nt * 16 + m;

    // loop-invariant slot metadata: s = wmma step (0..2), v = element (0..1)
    int  aoff[3][2];   // clamped in-plane offset for A load
    bool okA [3][2];   // A element valid (tap in-bounds and not a pad column)
    int  boff[3][2];   // weight-row offset for B load (clamped; pad cols killed by A=0)
#pragma unroll
    for (int s = 0; s < 3; ++s) {
#pragma unroll
        for (int v = 0; v < 2; ++v) {
            int rem = 4 * s + 2 * kp + v;          // [0,11]; >=9 are pad columns
            int dh = rem / 3, dw = rem % 3;
            int ih = ih0 + dh, iw = iw0 + dw;
            okA[s][v] = (rem < 9) && ((unsigned)ih < (unsigned)Hin)
                                  && ((unsigned)iw < (unsigned)Win);
            int ihc = min(max(ih, 0), Hin - 1);
            int iwc = min(max(iw, 0), Win - 1);
            aoff[s][v] = ihc * Win + iwc;
            boff[s][v] = oc * K9 + min(rem, 8);
        }
    }

    const float* xb = x + (size_t)b * Cin * plane;
    v8f acc = {};
    for (int ic = 0; ic < Cin; ++ic) {
        const float* xp = xb + (size_t)ic * plane;   // uniform base, per-lane offset
        const float* wp = w  + ic * 9;               // uniform base, per-lane offset
#pragma unroll
        for (int s = 0; s < 3; ++s) {
            v2f a, bm;
#pragma unroll
            for (int v = 0; v < 2; ++v) {
                float t = xp[aoff[s][v]];
                a[v]  = okA[s][v] ? t : 0.f;
                bm[v] = wp[boff[s][v]];
            }
            acc = __builtin_amdgcn_wmma_f32_16x16x4_f32(
                false, a, false, bm, (short)0, acc, false, false);
        }
    }

    // epilogue: bias + LeakyReLU(0.2); y[b][oc][pos] contiguous in pos
    int N = lane & 15, Mb = kp * 8;
    int oc2 = nt * 16 + N;
    float bia = bias[oc2];
    float* yb = y + ((size_t)b * Cout + oc2) * P;
#pragma unroll
    for (int r2 = 0; r2 < 8; ++r2) {
        float val = acc[r2] + bia;
        val = val >= 0.f ? val : 0.2f * val;
        yb[mt * 16 + Mb + r2] = val;
    }
}

// ---------------- 3. instance norm (affine) in place ----------------
__global__ __launch_bounds__(256) void instnorm_kernel(
    float* __restrict__ y, const float* __restrict__ gamma,
    const float* __restrict__ beta, int C, int HW)
{
    int bc = blockIdx.x;           // b*C + c
    int c  = bc % C;
    float* p = y + (size_t)bc * HW;
    __shared__ float s1[256], s2[256];
    float sum = 0.f, sq = 0.f;
    for (int i = threadIdx.x; i < HW; i += 256) { float v = p[i]; sum += v; sq += v * v; }
    s1[threadIdx.x] = sum; s2[threadIdx.x] = sq;
    __syncthreads();
    for (int off = 128; off; off >>= 1) {
        if (threadIdx.x < off) {
            s1[threadIdx.x] += s1[threadIdx.x + off];
            s2[threadIdx.x] += s2[threadIdx.x + off];
        }
        __syncthreads();
    }
    float inv_n = 1.f / (float)HW;
    float mu  = s1[0] * inv_n;
    float var = s2[0] * inv_n - mu * mu;
    float sc  = rsqrtf(var + 1e-5f) * gamma[c];
    float sh  = beta[c] - mu * sc;
    for (int i = threadIdx.x; i < HW; i += 256) p[i] = p[i] * sc + sh;
}

// ---------------- 4. adaptive avg pool 8x8 -> 2x2, flatten to codes --------
__global__ __launch_bounds__(256) void pool_codes_kernel(
    const float* __restrict__ y5, float* __restrict__ codes)
{
    int o = blockIdx.x * 256 + threadIdx.x;         // [0, 4*512)
    if (o >= 2048) return;
    int b = o >> 9, oo = o & 511;
    int c = oo >> 2, p = (oo >> 1) & 1, q = oo & 1;
    const float* base = y5 + ((((size_t)b * 128 + c) * 8) + p * 4) * 8 + q * 4;
    float s = 0.f;
#pragma unroll
    for (int u = 0; u < 4; ++u)
#pragma unroll
        for (int v = 0; v < 4; ++v) s += base[u * 8 + v];
    codes[o] = s * (1.f / 16.f);
}

// ---------------- 5. FC: out[b,o] = codes[b,:]·W[o,:] + bias[o] -------------
__global__ __launch_bounds__(256) void fc_kernel(
    const float* __restrict__ codes, const float* __restrict__ W,
    const float* __restrict__ bias, float* __restrict__ out,
    float* __restrict__ out2, int N)
{
    int b = blockIdx.x / N, o = blockIdx.x % N;
    __shared__ float s[256];
    const float* cb = codes + (size_t)b * 512;
    const float* wr = W + (size_t)o * 512;
    float acc = 0.f;
    for (int k = threadIdx.x; k < 512; k += 256) acc += cb[k] * wr[k];
    s[threadIdx.x] = acc; __syncthreads();
    for (int off = 128; off; off >>= 1) {
        if (threadIdx.x < off) s[threadIdx.x] += s[threadIdx.x + off];
        __syncthreads();
    }
    if (threadIdx.x == 0) {
        float v = s[0] + bias[o];
        out[blockIdx.x] = v;
        if (out2) out2[blockIdx.x] = v;
    }
}

// ---------------- 6. LUT bank expansion: luts[b,j] = sum_r wts[b,r]*bw[j,r] -
__global__ __launch_bounds__(256) void lutgen_kernel(
    const float* __restrict__ wts, const float* __restrict__ bw,
    float* __restrict__ luts)
{
    int i = blockIdx.x * 256 + threadIdx.x;
    const int PERB = 3 * LUT_D3;
    if (i >= 4 * PERB) return;
    int b = i / PERB, j = i % PERB;
    const float* br = bw + (size_t)j * 3;
    luts[i] = wts[b * 3 + 0] * br[0] + wts[b * 3 + 1] * br[1] + wts[b * 3 + 2] * br[2];
}

// ---------------- 7. softmax + cumsum -> vertices (one wave per (b,c)) -----
__global__ __launch_bounds__(32) void vertices_kernel(
    const float* __restrict__ iv, float* __restrict__ vws, float* __restrict__ vout)
{
    int row  = blockIdx.x;          // b*3 + c
    int lane = threadIdx.x;
    float x = iv[row * 32 + lane];
    float m = x;
    for (int off = 16; off; off >>= 1) m = fmaxf(m, __shfl_xor(m, off, 32));
    float e = expf(x - m);
    float s = e;
    for (int off = 16; off; off >>= 1) s += __shfl_xor(s, off, 32);
    float v = e / s;
    for (int off = 1; off < 32; off <<= 1) {        // inclusive scan
        float t = __shfl_up(v, off, 32);
        if (lane >= off) v += t;
    }
    int base = row * 33;
    if (lane == 0) { vws[base] = 0.f; vout[base] = 0.f; }
    vws[base + 1 + lane] = v;
    vout[base + 1 + lane] = v;
}

// ---------------- 8. full-res AiLUT transform (bandwidth-bound) -------------
// one pixel: 3x branchless searchsorted + trilerp of 3 output channels
__device__ __forceinline__ void ailut_pixel(
    const float* __restrict__ sv, const float* __restrict__ L,
    float pr, float pg, float pb, float& o0, float& o1, float& o2)
{
    float pin[3] = { pr, pg, pb };
    float coord[3];
#pragma unroll
    for (int c = 0; c < 3; ++c) {
        const float* v = sv + c * LUT_D;
        float p = pin[c];
        // v[0]==0 <= p always; find last idx with v[idx] <= p (branchless)
        int idx = 0;
#pragma unroll
        for (int step = 32; step > 0; step >>= 1) {
            int j = idx + step;
            bool ok = (j <= 32) && (v[j] <= p);
            idx = ok ? j : idx;
        }
        int ub = min(idx + 1, 32);                  // clip(searchsorted_right, 1, 32)
        float vlo = v[ub - 1], vhi = v[ub];
        float fr  = (p - vlo) / (vhi - vlo + 1e-8f);
        float cd  = (float)(ub - 1) + fr;
        coord[c]  = fminf(fmaxf(cd, 0.f), 32.f);
    }
    int i0 = min(max((int)floorf(coord[0]), 0), 31);
    int j0 = min(max((int)floorf(coord[1]), 0), 31);
    int k0 = min(max((int)floorf(coord[2]), 0), 31);
    float fr = coord[0] - (float)i0;
    float fg = coord[1] - (float)j0;
    float fb = coord[2] - (float)k0;
    float w000 = (1.f - fr) * (1.f - fg) * (1.f - fb);
    float w001 = (1.f - fr) * (1.f - fg) * fb;
    float w010 = (1.f - fr) * fg * (1.f - fb);
    float w011 = (1.f - fr) * fg * fb;
    float w100 = fr * (1.f - fg) * (1.f - fb);
    float w101 = fr * (1.f - fg) * fb;
    float w110 = fr * fg * (1.f - fb);
    float w111 = fr * fg * fb;
    int base000 = (i0 * LUT_D + j0) * LUT_D + k0;   // strides: i->1089, j->33, k->1
    float* outp[3] = { &o0, &o1, &o2 };
#pragma unroll
    for (int o = 0; o < 3; ++o) {
        const float* Lo = L + o * LUT_D3 + base000;
        *outp[o] = Lo[0]    * w000 + Lo[1]    * w001 + Lo[33]   * w010 + Lo[34]   * w011
                 + Lo[1089] * w100 + Lo[1090] * w101 + Lo[1122] * w110 + Lo[1123] * w111;
    }
}

__global__ __launch_bounds__(256) void ailut_kernel(
    const v4f* __restrict__ img4, const float* __restrict__ luts,
    const float* __restrict__ verts, v4f* __restrict__ out4)
{
    __shared__ float sv[3 * LUT_D];
    const int HW4 = (1024 * 1024) / 4;          // float4 per plane
    const int blocksPerB = HW4 / 256;           // 1024
    int b   = blockIdx.x / blocksPerB;
    int pix = (blockIdx.x % blocksPerB) * 256 + threadIdx.x;
    if (threadIdx.x < 3 * LUT_D) sv[threadIdx.x] = verts[b * 99 + threadIdx.x];
    __syncthreads();

    v4f pr = img4[((size_t)b * 3 + 0) * HW4 + pix];
    v4f pg = img4[((size_t)b * 3 + 1) * HW4 + pix];
    v4f pb = img4[((size_t)b * 3 + 2) * HW4 + pix];
    const float* L = luts + (size_t)b * 3 * LUT_D3;

    v4f r0, r1, r2;
#pragma unroll
    for (int e = 0; e < 4; ++e) {
        float o0, o1, o2;
        ailut_pixel(sv, L, pr[e], pg[e], pb[e], o0, o1, o2);
        r0[e] = fminf(fmaxf(o0, 0.f), 1.f);
        r1[e] = fminf(fmaxf(o1, 0.f), 1.f);
        r2[e] = fminf(fmaxf(o2, 0.f), 1.f);
    }
    out4[((size_t)b * 3 + 0) * HW4 + pix] = r0;
    out4[((size_t)b * 3 + 1) * HW4 + pix] = r1;
    out4[((size_t)b * 3 + 2) * HW4 + pix] = r2;
}

// ---------------- launcher --------------------------------------------------
extern "C" void kernel_launch(void* const* d_in, const int* in_sizes, int n_in,
                              void* d_out, int out_size, void* d_ws, size_t ws_size,
                              hipStream_t stream) {
    const float* lq = (const float*)d_in[0];
    const float *w1 = (const float*)d_in[1],  *b1 = (const float*)d_in[2];
    const float *g1 = (const float*)d_in[3],  *be1 = (const float*)d_in[4];
    const float *w2 = (const float*)d_in[5],  *b2 = (const float*)d_in[6];
    const float *g2 = (const float*)d_in[7],  *be2 = (const float*)d_in[8];
    const float *w3 = (const float*)d_in[9],  *b3 = (const float*)d_in[10];
    const float *g3 = (const float*)d_in[11], *be3 = (const float*)d_in[12];
    const float *w4 = (const float*)d_in[13], *b4 = (const float*)d_in[14];
    const float *g4 = (const float*)d_in[15], *be4 = (const float*)d_in[16];
    const float *w5 = (const float*)d_in[17], *b5 = (const float*)d_in[18];
    const float *lw = (const float*)d_in[19], *lb = (const float*)d_in[20];
    const float *bw = (const float*)d_in[21];
    const float *aw = (const float*)d_in[22], *ab = (const float*)d_in[23];

    float* out = (float*)d_out;
    float* ws  = (float*)d_ws;

    // workspace layout (floats)
    float* X0    = ws;                 // 4*3*256*256   = 786432
    float* Y1    = X0 + 786432;        // 4*16*128*128  = 1048576
    float* Y2    = Y1 + 1048576;       // 4*32*64*64    = 524288
    float* Y3    = Y2 + 524288;        // 4*64*32*32    = 262144
    float* Y4    = Y3 + 262144;        // 4*128*16*16   = 131072
    float* Y5    = Y4 + 131072;        // 4*128*8*8     = 32768
    float* CODES = Y5 + 32768;         // 4*512
    float* WTS   = CODES + 2048;       // 12 (pad 16)
    float* IV    = WTS + 16;           // 4*96
    float* VERT  = IV + 384;           // 4*3*33 (pad 400)
    float* LUTS  = VERT + 400;         // 4*3*35937 = 431244

    float* OUT_W = out + 12582912;     // weights [4,3]
    float* OUT_V = out + 12582924;     // vertices [4,3,33]

    // 1. resize
    resize4x_kernel<<<3072, 256, 0, stream>>>(lq, X0);

    // 2. backbone: conv(+bias+leaky) [+ instancenorm]; exact grids, 8 waves/blk
    conv_s2_wmma_kernel<<<512, 256, 0, stream>>>(X0, w1, b1, Y1, 3,   16,  256, 256, 7, 1024, 1);
    instnorm_kernel<<<64, 256, 0, stream>>>(Y1, g1, be1, 16, 128 * 128);

    conv_s2_wmma_kernel<<<256, 256, 0, stream>>>(Y1, w2, b2, Y2, 16,  32,  128, 128, 6, 256, 2);
    instnorm_kernel<<<128, 256, 0, stream>>>(Y2, g2, be2, 32, 64 * 64);

    conv_s2_wmma_kernel<<<128, 256, 0, stream>>>(Y2, w3, b3, Y3, 32,  64,  64,  64,  5, 64, 4);
    instnorm_kernel<<<256, 256, 0, stream>>>(Y3, g3, be3, 64, 32 * 32);

    conv_s2_wmma_kernel<<<64, 256, 0, stream>>>(Y3, w4, b4, Y4, 64,  128, 32,  32,  4, 16, 8);
    instnorm_kernel<<<512, 256, 0, stream>>>(Y4, g4, be4, 128, 16 * 16);

    conv_s2_wmma_kernel<<<16, 256, 0, stream>>>(Y4, w5, b5, Y5, 128, 128, 16,  16,  3, 4, 8);

    // 3. pool -> codes
    pool_codes_kernel<<<8, 256, 0, stream>>>(Y5, CODES);

    // 4. LUT generator + AdaInt heads
    fc_kernel<<<12, 256, 0, stream>>>(CODES, lw, lb, WTS, OUT_W, 3);
    lutgen_kernel<<<1685, 256, 0, stream>>>(WTS, bw, LUTS);
    fc_kernel<<<384, 256, 0, stream>>>(CODES, aw, ab, IV, nullptr, 96);
    vertices_kernel<<<12, 32, 0, stream>>>(IV, VERT, OUT_V);

    // 5. full-resolution LUT transform (one thread = 4 pixels, b128 I/O)
    ailut_kernel<<<4096, 256, 0, stream>>>((const v4f*)lq, LUTS, VERT, (v4f*)out);
}